// Baseline_GNN_79714593014136
// MI455X (gfx1250) — compile-verified
//
#include <hip/hip_runtime.h>
#include <cstddef>

typedef float v2f __attribute__((ext_vector_type(2)));
typedef float v8f __attribute__((ext_vector_type(8)));

#define DIN 64

// ---------------------------------------------------------------------------
// Zero a float buffer (graph-capture safe; atomics below need a clean base).
// ---------------------------------------------------------------------------
__global__ void gnn_zero_kernel(float* __restrict__ p, unsigned long long n) {
    unsigned long long i = (unsigned long long)blockIdx.x * blockDim.x + threadIdx.x;
    if (i < n) p[i] = 0.0f;
}

// ---------------------------------------------------------------------------
// One-time weight pack: WT[col*64+k] = W[k*dout+col], zero-padded to dout_pad
// columns; bias padded likewise. Makes the WMMA B-operand loads contiguous,
// unconditional b64 loads (no exec-mask juggling in the hot loop).
// ---------------------------------------------------------------------------
__global__ void gnn_pack_weights_kernel(const float* __restrict__ Wl,
                                        const float* __restrict__ bl,
                                        const float* __restrict__ Wr,
                                        float* __restrict__ WTl,
                                        float* __restrict__ WTr,
                                        float* __restrict__ blp,
                                        int dout, int dout_pad) {
    int tid = blockIdx.x * blockDim.x + threadIdx.x;     // over dout_pad * DIN
    int col = tid / DIN;
    int k   = tid % DIN;
    if (col >= dout_pad) return;
    bool cv  = (col < dout);
    WTl[(size_t)col * DIN + k] = cv ? Wl[(size_t)k * dout + col] : 0.0f;
    WTr[(size_t)col * DIN + k] = cv ? Wr[(size_t)k * dout + col] : 0.0f;
    if (k == 0) blp[col] = cv ? bl[col] : 0.0f;
}

// ---------------------------------------------------------------------------
// Edge scatter: agg[dst] += x[src] (and deg[dst] += 1 when do_deg).
// 16 lanes per edge; each lane moves one float4 of the 64-wide feature row.
// ---------------------------------------------------------------------------
__global__ void gnn_scatter_kernel(const float* __restrict__ x,
                                   const int* __restrict__ src,
                                   const int* __restrict__ dst,
                                   float* __restrict__ agg,
                                   float* __restrict__ deg,
                                   int E, int do_deg) {
    unsigned long long tid = (unsigned long long)blockIdx.x * blockDim.x + threadIdx.x;
    unsigned long long e  = tid >> 4;
    int f4 = (int)(tid & 15);
    if (e >= (unsigned long long)E) return;
    int s = src[e];
    int d = dst[e];
    const float4* xr = (const float4*)(x + (size_t)s * DIN);
    float4 v = xr[f4];
    float* ar = agg + (size_t)d * DIN + (size_t)f4 * 4;
    atomicAdd(ar + 0, v.x);
    atomicAdd(ar + 1, v.y);
    atomicAdd(ar + 2, v.z);
    atomicAdd(ar + 3, v.w);
    if (do_deg && f4 == 0) atomicAdd(deg + d, 1.0f);
}

// ---------------------------------------------------------------------------
// Fused SAGE dense step via fp32 WMMA, fully branch-free inner loop:
//   hout = [relu]( (agg/deg) @ Wl + hin @ Wr + blp )
// One wave computes a 16x16 tile; K-loop over DIN in steps of 4 issues two
// v_wmma_f32_16x16x4_f32 per step on a shared f32 accumulator. Weights come
// from the packed column-major buffers (contiguous b64 loads, padded cols are
// zero so no masking anywhere; padded outputs land in stride-64 scratch).
// ---------------------------------------------------------------------------
__global__ void __launch_bounds__(128)
gnn_dense_wmma_kernel(const float* __restrict__ agg,
                      const float* __restrict__ deg,
                      const float* __restrict__ hin,
                      const float* __restrict__ WTl,   // [dout_pad][64]
                      const float* __restrict__ blp,   // [dout_pad]
                      const float* __restrict__ WTr,   // [dout_pad][64]
                      float* __restrict__ hout,        // stride 64
                      int do_relu, int n_mtiles) {
    const int lane  = threadIdx.x;                 // 0..31 (wave32)
    const int mtile = blockIdx.x * blockDim.y + threadIdx.y;
    const int ntile = blockIdx.y;
    if (mtile >= n_mtiles) return;                 // wave-uniform exit

    const int mrow  = mtile * 16 + (lane & 15);    // A row (M = lane&15)
    const int khalf = (lane >> 4) * 2;             // K sub-block from lane half
    const int col   = ntile * 16 + (lane & 15);    // B/C column (N = lane&15)

    const float rdeg = 1.0f / fmaxf(deg[mrow], 1.0f);
    const float* aggr = agg + (size_t)mrow * DIN + khalf;
    const float* hinr = hin + (size_t)mrow * DIN + khalf;
    const float* wl   = WTl + (size_t)col  * DIN + khalf;
    const float* wr   = WTr + (size_t)col  * DIN + khalf;

    v8f c = {};
    #pragma unroll
    for (int k0 = 0; k0 < DIN; k0 += 4) {
        // A (16x4 f32): lane holds M=lane&15, K = khalf + vgpr — b64 loads
        v2f aAgg = *(const v2f*)(aggr + k0);
        aAgg.x *= rdeg; aAgg.y *= rdeg;
        v2f aX   = *(const v2f*)(hinr + k0);
        // B (4x16 f32): lane holds N=lane&15, K = khalf + vgpr — b64 loads
        v2f bL   = *(const v2f*)(wl + k0);
        v2f bR   = *(const v2f*)(wr + k0);
        c = __builtin_amdgcn_wmma_f32_16x16x4_f32(false, aAgg, false, bL,
                                                  (short)0, c, false, false);
        c = __builtin_amdgcn_wmma_f32_16x16x4_f32(false, aX,   false, bR,
                                                  (short)0, c, false, false);
    }

    const float bv = blp[col];
    // C/D layout: VGPR r -> lanes 0-15: M=r, lanes 16-31: M=r+8; N = lane&15
    float* outp = hout + (size_t)(mtile * 16 + ((lane >> 4) * 8)) * DIN + col;
    #pragma unroll
    for (int r = 0; r < 8; ++r) {
        float v = c[r] + bv;
        if (do_relu) v = fmaxf(v, 0.0f);
        outp[(size_t)r * DIN] = v;
    }
}

// ---------------------------------------------------------------------------
// Row-wise log_softmax over C=40 logits (row stride 64 in scratch).
// One wave per node; wave32 xor-shuffle reductions.
// ---------------------------------------------------------------------------
__global__ void gnn_logsoftmax_kernel(const float* __restrict__ logits,
                                      float* __restrict__ out,
                                      int n_nodes, int C, int stride) {
    const int lane = threadIdx.x;                  // 0..31
    const int node = blockIdx.x * blockDim.y + threadIdx.y;
    if (node >= n_nodes) return;
    const float* rowp = logits + (size_t)node * stride;
    const float NEG_INF = -__builtin_inff();
    float v0 = (lane      < C) ? rowp[lane]      : NEG_INF;
    float v1 = (lane + 32 < C) ? rowp[lane + 32] : NEG_INF;
    float m = fmaxf(v0, v1);
    #pragma unroll
    for (int off = 16; off > 0; off >>= 1) m = fmaxf(m, __shfl_xor(m, off, 32));
    float e0 = (lane      < C) ? expf(v0 - m) : 0.0f;
    float e1 = (lane + 32 < C) ? expf(v1 - m) : 0.0f;
    float s = e0 + e1;
    #pragma unroll
    for (int off = 16; off > 0; off >>= 1) s += __shfl_xor(s, off, 32);
    const float ls = logf(s);
    if (lane      < C) out[(size_t)node * C + lane]      = v0 - m - ls;
    if (lane + 32 < C) out[(size_t)node * C + lane + 32] = v1 - m - ls;
}

// ---------------------------------------------------------------------------
// Host-side orchestration (all launches on `stream`; graph-capture safe).
// ---------------------------------------------------------------------------
static inline void launch_dense(const float* agg, const float* deg, const float* hin,
                                const float* WTl, const float* blp, const float* WTr,
                                float* hout, int dout_pad, int do_relu,
                                int N, hipStream_t stream) {
    const int n_mtiles = (N + 15) / 16;
    dim3 block(32, 4);
    dim3 grid((n_mtiles + 3) / 4, dout_pad / 16);
    gnn_dense_wmma_kernel<<<grid, block, 0, stream>>>(agg, deg, hin, WTl, blp, WTr,
                                                      hout, do_relu, n_mtiles);
}

extern "C" void kernel_launch(void* const* d_in, const int* in_sizes, int n_in,
                              void* d_out, int out_size, void* d_ws, size_t ws_size,
                              hipStream_t stream) {
    const float* x   = (const float*)d_in[0];
    const int*   ei  = (const int*)  d_in[1];
    const float* Wl0 = (const float*)d_in[2];
    const float* bl0 = (const float*)d_in[3];
    const float* Wr0 = (const float*)d_in[4];
    const float* Wl1 = (const float*)d_in[5];
    const float* bl1 = (const float*)d_in[6];
    const float* Wr1 = (const float*)d_in[7];
    const float* Wl2 = (const float*)d_in[8];
    const float* bl2 = (const float*)d_in[9];
    const float* Wr2 = (const float*)d_in[10];

    const int N = in_sizes[0] / DIN;     // 100000
    const int E = in_sizes[1] / 2;       // 1600000
    const int* src = ei;
    const int* dst = ei + E;

    const int DP0 = 64, DP1 = 64, DP2 = 48;   // padded dout per layer

    // Workspace (floats):
    //   deg[N] | agg[N*64] | h1[N*64] | h2[N*64] |
    //   WTl0[64*64] WTr0[64*64] blp0[64] | WTl1 .. | WTl2[48*64] WTr2[48*64] blp2[48]
    float* deg  = (float*)d_ws;
    float* agg  = deg + N;
    float* h1   = agg + (size_t)N * DIN;
    float* h2   = h1  + (size_t)N * DIN;
    float* WTl0 = h2  + (size_t)N * DIN;
    float* WTr0 = WTl0 + DP0 * DIN;
    float* blp0 = WTr0 + DP0 * DIN;
    float* WTl1 = blp0 + DP0;
    float* WTr1 = WTl1 + DP1 * DIN;
    float* blp1 = WTr1 + DP1 * DIN;
    float* WTl2 = blp1 + DP1;
    float* WTr2 = WTl2 + DP2 * DIN;
    float* blp2 = WTr2 + DP2 * DIN;

    const unsigned long long n_da = (unsigned long long)N * (DIN + 1);
    const unsigned long long n_a  = (unsigned long long)N * DIN;
    const unsigned long long scat_threads = (unsigned long long)E * 16ull;
    const int ZB = 256;

    // ---- Pack weights (tiny one-time pre-pass) ----
    gnn_pack_weights_kernel<<<(DP0 * DIN + ZB - 1) / ZB, ZB, 0, stream>>>(
        Wl0, bl0, Wr0, WTl0, WTr0, blp0, 64, DP0);
    gnn_pack_weights_kernel<<<(DP1 * DIN + ZB - 1) / ZB, ZB, 0, stream>>>(
        Wl1, bl1, Wr1, WTl1, WTr1, blp1, 64, DP1);
    gnn_pack_weights_kernel<<<(DP2 * DIN + ZB - 1) / ZB, ZB, 0, stream>>>(
        Wl2, bl2, Wr2, WTl2, WTr2, blp2, 40, DP2);

    // ---- Layer 0 ----
    gnn_zero_kernel<<<(unsigned)((n_da + ZB - 1) / ZB), ZB, 0, stream>>>(deg, n_da);
    gnn_scatter_kernel<<<(unsigned)((scat_threads + ZB - 1) / ZB), ZB, 0, stream>>>(
        x, src, dst, agg, deg, E, 1);
    launch_dense(agg, deg, x, WTl0, blp0, WTr0, h1, DP0, 1, N, stream);

    // ---- Layer 1 ----
    gnn_zero_kernel<<<(unsigned)((n_a + ZB - 1) / ZB), ZB, 0, stream>>>(agg, n_a);
    gnn_scatter_kernel<<<(unsigned)((scat_threads + ZB - 1) / ZB), ZB, 0, stream>>>(
        h1, src, dst, agg, deg, E, 0);
    launch_dense(agg, deg, h1, WTl1, blp1, WTr1, h2, DP1, 1, N, stream);

    // ---- Layer 2 (logits into h1, stride 64; padded cols 40..47 are zeroed W) ----
    gnn_zero_kernel<<<(unsigned)((n_a + ZB - 1) / ZB), ZB, 0, stream>>>(agg, n_a);
    gnn_scatter_kernel<<<(unsigned)((scat_threads + ZB - 1) / ZB), ZB, 0, stream>>>(
        h2, src, dst, agg, deg, E, 0);
    launch_dense(agg, deg, h2, WTl2, blp2, WTr2, h1, DP2, 0, N, stream);

    // ---- log_softmax over the 40 logits ----
    {
        dim3 block(32, 8);
        dim3 grid((N + 7) / 8);
        gnn_logsoftmax_kernel<<<grid, block, 0, stream>>>(h1, (float*)d_out, N, 40, 64);
    }
}